// MultiModalGCNVAE_89850715832384
// MI455X (gfx1250) — compile-verified
//
#include <hip/hip_runtime.h>
#include <math.h>

// Problem constants (match reference)
#define N_NODES  100000
#define E_EDGES  3200000
#define DIM_IN   100
#define H_DIM    64
#define L_DIM    64
#define NODE_NUM 100

typedef __attribute__((ext_vector_type(2))) float v2f;
typedef __attribute__((ext_vector_type(8))) float v8f;

__device__ __forceinline__ v8f wmma4(v2f a, v2f b, v8f c) {
  // D = A(16x4,f32) * B(4x16,f32) + C(16x16,f32)  -> v_wmma_f32_16x16x4_f32
  return __builtin_amdgcn_wmma_f32_16x16x4_f32(false, a, false, b, (short)0, c,
                                               false, false);
}

// ---------------- utility kernels ----------------

__global__ void zero_kernel(float* __restrict__ p, long long n) {
  long long i = (long long)blockIdx.x * blockDim.x + threadIdx.x;
  if (i < n) p[i] = 0.0f;
}

__global__ void deg_kernel(const int* __restrict__ ei, int* __restrict__ deg, int E) {
  int e = blockIdx.x * blockDim.x + threadIdx.x;
  if (e < E) atomicAdd(&deg[ei[E + e]], 1);   // dst row of edge_index
}

__global__ void dinv_kernel(const int* __restrict__ deg, float* __restrict__ dinv, int n) {
  int i = blockIdx.x * blockDim.x + threadIdx.x;
  if (i < n) dinv[i] = rsqrtf((float)(deg[i] + 1));  // +1 for self loop; deg>=1 always
}

// Gather h[src]*norm, scatter-add into agg[dst]. 16 lanes per edge, float4 each:
// coalesced 256B gather per edge; h/agg are L2-resident (25.6MB each) so the
// random f32 atomic adds execute at the L2 atomic units, not HBM.
__global__ void edge_scatter_kernel(const int* __restrict__ ei,
                                    const float* __restrict__ dinv,
                                    const float* __restrict__ h,
                                    float* __restrict__ agg, int E) {
  long long gid = (long long)blockIdx.x * blockDim.x + threadIdx.x;
  if (gid >= (long long)E * 16) return;
  int e    = (int)(gid >> 4);
  int part = (int)(gid & 15);
  int src = ei[e];
  int dst = ei[E + e];
  float norm = dinv[src] * dinv[dst];
  const float4* hp = (const float4*)(h + (size_t)src * H_DIM);
  float4 v = hp[part];
  float* ap = agg + (size_t)dst * H_DIM + part * 4;
  atomicAdd(ap + 0, v.x * norm);
  atomicAdd(ap + 1, v.y * norm);
  atomicAdd(ap + 2, v.z * norm);
  atomicAdd(ap + 3, v.w * norm);
}

// out[i][c] = act( agg[i][c] + h[i][c]*dinv[i]^2 + bias[c] )  (self-loop fused)
__global__ void post_kernel(const float* __restrict__ agg, const float* __restrict__ h,
                            const float* __restrict__ dinv, const float* __restrict__ bias,
                            float* __restrict__ out, int n, int ldOut, int do_relu) {
  long long gid = (long long)blockIdx.x * blockDim.x + threadIdx.x;
  if (gid >= (long long)n * H_DIM) return;
  int i = (int)(gid >> 6);
  int c = (int)(gid & 63);
  float di = dinv[i];
  float v = agg[gid] + h[gid] * (di * di) + bias[c];
  if (do_relu) v = fmaxf(v, 0.0f);
  out[(size_t)i * ldOut + c] = v;
}

// z = mu + eps * exp(0.5*logvar)
__global__ void reparam_kernel(const float* __restrict__ mu, const float* __restrict__ lv,
                               const float* __restrict__ eps, float* __restrict__ z,
                               long long n) {
  long long i = (long long)blockIdx.x * blockDim.x + threadIdx.x;
  if (i < n) z[i] = fmaf(eps[i], expf(0.5f * lv[i]), mu[i]);
}

// ---------------- WMMA GEMMs (f32 16x16x4) ----------------
// One wave computes a 16-row x 64-col tile of C = A[MxK] @ B[Kx64] (+bias).
// Fragment layouts per CDNA5 ISA 7.12.2 (32-bit A 16x4, mirrored B, 8-VGPR C).
__global__ void gemm_n64_wmma(const float* __restrict__ A, int lda,
                              const float* __restrict__ B,      // [K x 64], ldb = 64
                              const float* __restrict__ bias,   // nullptr -> none
                              float* __restrict__ C, int ldc,
                              int M, int K) {
  int wave    = blockIdx.x * (blockDim.x >> 5) + (threadIdx.x >> 5);
  int rowBase = wave << 4;
  if (rowBase >= M) return;            // wave-uniform: EXEC stays all-ones for WMMA
  int lane = threadIdx.x & 31;
  int half = lane >> 4;                // 0: K pair {0,1}; 1: K pair {2,3}
  int r    = lane & 15;

  v8f acc0 = {}, acc1 = {}, acc2 = {}, acc3 = {};
  const float* arow = A + (size_t)(rowBase + r) * lda + 2 * half;

  for (int k0 = 0; k0 < K; k0 += 4) {
    v2f a;
    a.x = arow[k0];
    a.y = arow[k0 + 1];
    const float* b0 = B + (size_t)(k0 + 2 * half) * 64 + r;
    const float* b1 = b0 + 64;
    v2f b;
    b.x = b0[0];  b.y = b1[0];  acc0 = wmma4(a, b, acc0);
    b.x = b0[16]; b.y = b1[16]; acc1 = wmma4(a, b, acc1);
    b.x = b0[32]; b.y = b1[32]; acc2 = wmma4(a, b, acc2);
    b.x = b0[48]; b.y = b1[48]; acc3 = wmma4(a, b, acc3);
  }

  float bb0 = 0.f, bb1 = 0.f, bb2 = 0.f, bb3 = 0.f;
  if (bias) { bb0 = bias[r]; bb1 = bias[16 + r]; bb2 = bias[32 + r]; bb3 = bias[48 + r]; }

#pragma unroll
  for (int v = 0; v < 8; ++v) {
    size_t row = (size_t)rowBase + v + 8 * half;   // C layout: M = v + 8*half
    float* crow = C + row * (size_t)ldc;
    crow[r]      = acc0[v] + bb0;
    crow[16 + r] = acc1[v] + bb1;
    crow[32 + r] = acc2[v] + bb2;
    crow[48 + r] = acc3[v] + bb3;
  }
}

// Decoder: Out[Mx100] = tanh(Z[Mx64] @ Wd[64x100] + bd). 7 col tiles (112), masked.
__global__ void decoder_wmma(const float* __restrict__ Z, int ldz,
                             const float* __restrict__ Wd,   // [64 x 100]
                             const float* __restrict__ bd,
                             float* __restrict__ Out,        // [M x 100]
                             int M) {
  int wave    = blockIdx.x * (blockDim.x >> 5) + (threadIdx.x >> 5);
  int rowBase = wave << 4;
  if (rowBase >= M) return;
  int lane = threadIdx.x & 31;
  int half = lane >> 4;
  int r    = lane & 15;

  v8f acc[7] = {};
  const float* arow = Z + (size_t)(rowBase + r) * ldz + 2 * half;

  for (int k0 = 0; k0 < 64; k0 += 4) {
    v2f a;
    a.x = arow[k0];
    a.y = arow[k0 + 1];
    const float* b0 = Wd + (size_t)(k0 + 2 * half) * NODE_NUM;
    const float* b1 = b0 + NODE_NUM;
#pragma unroll
    for (int t = 0; t < 7; ++t) {
      int col = t * 16 + r;
      v2f b;
      b.x = (col < NODE_NUM) ? b0[col] : 0.0f;   // select, not branch: EXEC intact
      b.y = (col < NODE_NUM) ? b1[col] : 0.0f;
      acc[t] = wmma4(a, b, acc[t]);
    }
  }

#pragma unroll
  for (int t = 0; t < 7; ++t) {
    int col = t * 16 + r;
    if (col < NODE_NUM) {
      float bv = bd[col];
#pragma unroll
      for (int v = 0; v < 8; ++v) {
        size_t row = (size_t)rowBase + v + 8 * half;
        Out[row * NODE_NUM + col] = tanhf(acc[t][v] + bv);
      }
    }
  }
}

// ---------------- launch ----------------

extern "C" void kernel_launch(void* const* d_in, const int* in_sizes, int n_in,
                              void* d_out, int out_size, void* d_ws, size_t ws_size,
                              hipStream_t stream) {
  (void)in_sizes; (void)n_in; (void)out_size; (void)ws_size;
  const int N = N_NODES, E = E_EDGES;

  const float* x[3]  = {(const float*)d_in[0], (const float*)d_in[1], (const float*)d_in[2]};
  const int*   ei[3] = {(const int*)d_in[3], (const int*)d_in[4], (const int*)d_in[5]};
  const float* eps   = (const float*)d_in[6];
  const float* W1[3] = {(const float*)d_in[7],  (const float*)d_in[11], (const float*)d_in[15]};
  const float* b1[3] = {(const float*)d_in[8],  (const float*)d_in[12], (const float*)d_in[16]};
  const float* W2[3] = {(const float*)d_in[9],  (const float*)d_in[13], (const float*)d_in[17]};
  const float* b2[3] = {(const float*)d_in[10], (const float*)d_in[14], (const float*)d_in[18]};
  const float* Wmu = (const float*)d_in[19]; const float* bmu = (const float*)d_in[20];
  const float* Wlv = (const float*)d_in[21]; const float* blv = (const float*)d_in[22];
  const float* Wd[3] = {(const float*)d_in[23], (const float*)d_in[25], (const float*)d_in[27]};
  const float* bd[3] = {(const float*)d_in[24], (const float*)d_in[26], (const float*)d_in[28]};

  // d_out layout: rs | ri | rd | mu | logvar
  float* out    = (float*)d_out;
  float* mu_out = out + (size_t)3 * N * NODE_NUM;
  float* lv_out = mu_out + (size_t)N * 3 * L_DIM;

  // workspace partition (~123 MiB)
  float* h_all  = (float*)d_ws;                    // N * 192  (later reused for z)
  float* hbuf   = h_all + (size_t)N * 192;         // N * 64
  float* aggbuf = hbuf  + (size_t)N * 64;          // N * 64
  int*   deg    = (int*)(aggbuf + (size_t)N * 64); // N
  float* dinv   = (float*)(deg + N);               // N

  const int TB = 256;
  auto blks = [](long long n, int tb) { return (unsigned)((n + tb - 1) / tb); };
  const unsigned gN    = blks(N, TB);
  const unsigned gE    = blks(E, TB);
  const unsigned gN64  = blks((long long)N * 64, TB);
  const unsigned gE16  = blks((long long)E * 16, TB);
  const unsigned gGemN = blks((long long)(N / 16) * 32, TB);       // one wave / 16 rows

  for (int m = 0; m < 3; ++m) {
    // degree + normalization (shared by both layers)
    zero_kernel<<<gN, TB, 0, stream>>>((float*)deg, N);
    deg_kernel<<<gE, TB, 0, stream>>>(ei[m], deg, E);
    dinv_kernel<<<gN, TB, 0, stream>>>(deg, dinv, N);

    // layer 1: h1 = x @ W1 ; agg ; r1 = relu(agg + selfloop + b1) (in-place in aggbuf)
    gemm_n64_wmma<<<gGemN, TB, 0, stream>>>(x[m], DIM_IN, W1[m], nullptr, hbuf, 64, N, DIM_IN);
    zero_kernel<<<gN64, TB, 0, stream>>>(aggbuf, (long long)N * 64);
    edge_scatter_kernel<<<gE16, TB, 0, stream>>>(ei[m], dinv, hbuf, aggbuf, E);
    post_kernel<<<gN64, TB, 0, stream>>>(aggbuf, hbuf, dinv, b1[m], aggbuf, N, 64, 1);

    // layer 2: h2 = r1 @ W2 ; agg ; h_m = agg + selfloop + b2 -> h_all[:, m, :]
    gemm_n64_wmma<<<gGemN, TB, 0, stream>>>(aggbuf, 64, W2[m], nullptr, hbuf, 64, N, 64);
    zero_kernel<<<gN64, TB, 0, stream>>>(aggbuf, (long long)N * 64);
    edge_scatter_kernel<<<gE16, TB, 0, stream>>>(ei[m], dinv, hbuf, aggbuf, E);
    post_kernel<<<gN64, TB, 0, stream>>>(aggbuf, hbuf, dinv, b2[m], h_all + m * 64, N, 192, 0);
  }

  // mu / logvar: [3N x 64] @ [64 x 64] + bias, straight into d_out
  const int M3 = 3 * N;
  const unsigned gGemM3 = blks((long long)(M3 / 16) * 32, TB);
  gemm_n64_wmma<<<gGemM3, TB, 0, stream>>>(h_all, 64, Wmu, bmu, mu_out, 64, M3, 64);
  gemm_n64_wmma<<<gGemM3, TB, 0, stream>>>(h_all, 64, Wlv, blv, lv_out, 64, M3, 64);

  // z = mu + eps*exp(0.5*logvar), overwrite h_all (h no longer needed)
  reparam_kernel<<<blks((long long)M3 * 64, TB), TB, 0, stream>>>(
      mu_out, lv_out, eps, h_all, (long long)M3 * 64);

  // decoders: tanh(z_m @ Wd_m + bd_m) -> rs/ri/rd
  for (int m = 0; m < 3; ++m) {
    decoder_wmma<<<gGemN, TB, 0, stream>>>(h_all + m * 64, 192, Wd[m], bd[m],
                                           out + (size_t)m * N * NODE_NUM, N);
  }
}